// GlobalModelMulti_39444979647204
// MI455X (gfx1250) — compile-verified
//
#include <hip/hip_runtime.h>
#include <math.h>

#define H       128
#define MH      64
#define KC      (5 * H)     // 640
#define NPB     512         // nodes (rows) per reduction block
#define LN_EPS  1e-5f

typedef __attribute__((ext_vector_type(2))) float v2f;
typedef __attribute__((ext_vector_type(4))) float v4f;
typedef __attribute__((ext_vector_type(8))) float v8f;

// ---------------------------------------------------------------------------
// Workspace init: sum=0, sumsq=0, min=+inf, max=-inf, cnt=0
// ---------------------------------------------------------------------------
__global__ void init_ws_kernel(float* __restrict__ s, float* __restrict__ sq,
                               float* __restrict__ mi, float* __restrict__ ma,
                               float* __restrict__ cnt, int nB) {
    int i = blockIdx.x * blockDim.x + threadIdx.x;
    if (i < nB * H) {
        s[i]  = 0.f;
        sq[i] = 0.f;
        mi[i] = __builtin_inff();
        ma[i] = -__builtin_inff();
    }
    if (i < nB) cnt[i] = 0.f;
}

// ---------------------------------------------------------------------------
// Segmented reduction over sorted batch ids.
// Block = 128 threads = 4 row-substreams (one per wave32) x 32 column groups.
// Each lane loads a v4f (GLOBAL_LOAD_B128, 512B per wave request) with a
// non-temporal hint (x is 1GB single-use, > 192MB L2). Register accumulation;
// atomic flush only at segment boundaries (~1953-row average segments).
// The boundary branch is wave-uniform (all 32 lanes of a substream read the
// same batch[n]).
// ---------------------------------------------------------------------------
__global__ void __launch_bounds__(128)
seg_reduce_kernel(const float* __restrict__ x, const int* __restrict__ batch,
                  float* __restrict__ s, float* __restrict__ sq,
                  float* __restrict__ mi, float* __restrict__ ma,
                  float* __restrict__ cnt, int n_nodes) {
    const int tid  = threadIdx.x;
    const int cg   = tid & 31;   // column group -> columns 4*cg .. 4*cg+3
    const int rsub = tid >> 5;   // row substream 0..3 (== wave id)
    const int c0   = cg * 4;

    const int base  = blockIdx.x * NPB;
    const int start = base + rsub;
    if (start >= n_nodes) return;
    int end = base + NPB;
    if (end > n_nodes) end = n_nodes;

    float a0 = 0.f, a1 = 0.f, a2 = 0.f, a3 = 0.f;       // sum
    float q0 = 0.f, q1 = 0.f, q2 = 0.f, q3 = 0.f;       // sum of squares
    float n0, n1, n2, n3, m0, m1, m2, m3;               // min / max
    n0 = n1 = n2 = n3 = __builtin_inff();
    m0 = m1 = m2 = m3 = -__builtin_inff();
    int run = 0;
    int cur = batch[start];

    for (int n = start; n < end; n += 4) {
        int b = batch[n];
        if (b != cur) {
            float* sp = &s[cur * H + c0];
            float* qp = &sq[cur * H + c0];
            float* ip = &mi[cur * H + c0];
            float* ap = &ma[cur * H + c0];
            atomicAdd(sp + 0, a0); atomicAdd(sp + 1, a1);
            atomicAdd(sp + 2, a2); atomicAdd(sp + 3, a3);
            atomicAdd(qp + 0, q0); atomicAdd(qp + 1, q1);
            atomicAdd(qp + 2, q2); atomicAdd(qp + 3, q3);
            atomicMin(ip + 0, n0); atomicMin(ip + 1, n1);
            atomicMin(ip + 2, n2); atomicMin(ip + 3, n3);
            atomicMax(ap + 0, m0); atomicMax(ap + 1, m1);
            atomicMax(ap + 2, m2); atomicMax(ap + 3, m3);
            if (cg == 0) atomicAdd(&cnt[cur], (float)run);
            a0 = a1 = a2 = a3 = 0.f;
            q0 = q1 = q2 = q3 = 0.f;
            n0 = n1 = n2 = n3 = __builtin_inff();
            m0 = m1 = m2 = m3 = -__builtin_inff();
            run = 0;
            cur = b;
        }
        v4f v = __builtin_nontemporal_load(
            (const v4f*)&x[(size_t)n * H + c0]);
        a0 += v.x; a1 += v.y; a2 += v.z; a3 += v.w;
        q0 += v.x * v.x; q1 += v.y * v.y;
        q2 += v.z * v.z; q3 += v.w * v.w;
        n0 = fminf(n0, v.x); n1 = fminf(n1, v.y);
        n2 = fminf(n2, v.z); n3 = fminf(n3, v.w);
        m0 = fmaxf(m0, v.x); m1 = fmaxf(m1, v.y);
        m2 = fmaxf(m2, v.z); m3 = fmaxf(m3, v.w);
        ++run;
    }
    {
        float* sp = &s[cur * H + c0];
        float* qp = &sq[cur * H + c0];
        float* ip = &mi[cur * H + c0];
        float* ap = &ma[cur * H + c0];
        atomicAdd(sp + 0, a0); atomicAdd(sp + 1, a1);
        atomicAdd(sp + 2, a2); atomicAdd(sp + 3, a3);
        atomicAdd(qp + 0, q0); atomicAdd(qp + 1, q1);
        atomicAdd(qp + 2, q2); atomicAdd(qp + 3, q3);
        atomicMin(ip + 0, n0); atomicMin(ip + 1, n1);
        atomicMin(ip + 2, n2); atomicMin(ip + 3, n3);
        atomicMax(ap + 0, m0); atomicMax(ap + 1, m1);
        atomicMax(ap + 2, m2); atomicMax(ap + 3, m3);
        if (cg == 0) atomicAdd(&cnt[cur], (float)run);
    }
}

// ---------------------------------------------------------------------------
// concat[b, 0:640] = [u, s, mi, ma, var]  where var = sq/cnt - (s/cnt)^2
// ---------------------------------------------------------------------------
__global__ void build_concat_kernel(const float* __restrict__ u,
                                    const float* __restrict__ s,
                                    const float* __restrict__ sq,
                                    const float* __restrict__ mi,
                                    const float* __restrict__ ma,
                                    const float* __restrict__ cnt,
                                    float* __restrict__ concat, int nB) {
    int i = blockIdx.x * blockDim.x + threadIdx.x;
    if (i >= nB * KC) return;
    int b = i / KC;
    int j = i % KC;
    float v;
    if (j < H)          v = u[b * H + j];
    else if (j < 2 * H) v = s[b * H + (j - H)];
    else if (j < 3 * H) v = mi[b * H + (j - 2 * H)];
    else if (j < 4 * H) v = ma[b * H + (j - 3 * H)];
    else {
        int jj = j - 4 * H;
        float cv = fmaxf(cnt[b], 1.f);
        float me = s[b * H + jj] / cv;
        v = sq[b * H + jj] / cv - me * me;
    }
    concat[i] = v;
}

// ---------------------------------------------------------------------------
// MLP via V_WMMA_F32_16X16X4_F32. One wave (32 lanes) per 16-row tile.
// A-frag (16x4 f32): a.x/a.y = A[lane&15][k + 2*(lane>>4) + {0,1}]
// B-frag (4x16 f32): b.x/b.y = B[k + 2*(lane>>4) + {0,1}][lane&15]
// C/D (16x16 f32, 8 VGPRs): elem v -> row (v + 8*(lane>>4)), col (lane&15)
// ---------------------------------------------------------------------------
__global__ void __launch_bounds__(32)
mlp_kernel(const float* __restrict__ concat, const float* __restrict__ u,
           const float* __restrict__ W0, const float* __restrict__ b0,
           const float* __restrict__ W1, const float* __restrict__ b1,
           const float* __restrict__ W2, const float* __restrict__ b2,
           const float* __restrict__ lng, const float* __restrict__ lnb,
           const float* __restrict__ W3, const float* __restrict__ b3,
           float* __restrict__ out) {
    __shared__ float hbuf[16][MH + 4];

    const int lane  = threadIdx.x;
    const int rbase = blockIdx.x * 16;
    const int mrow  = lane & 15;
    const int hi    = lane >> 4;       // 0 or 1
    const int koff  = hi * 2;
    const int colN  = lane & 15;

    // ---- layer 0: [16 x 640] @ [640 x 64] + b0, relu ----
    v8f acc[4];
    for (int t = 0; t < 4; ++t) {
        float bv = b0[t * 16 + colN];
        for (int v = 0; v < 8; ++v) acc[t][v] = bv;
    }
    for (int kk = 0; kk < KC; kk += 4) {
        int k0 = kk + koff;
        v2f a = *(const v2f*)(&concat[(size_t)(rbase + mrow) * KC + k0]);
        for (int t = 0; t < 4; ++t) {
            v2f bb;
            bb.x = W0[(size_t)k0 * MH + t * 16 + colN];
            bb.y = W0[(size_t)(k0 + 1) * MH + t * 16 + colN];
            acc[t] = __builtin_amdgcn_wmma_f32_16x16x4_f32(
                false, a, false, bb, (short)0, acc[t], false, false);
        }
    }
    for (int t = 0; t < 4; ++t)
        for (int v = 0; v < 8; ++v)
            hbuf[v + hi * 8][t * 16 + colN] = fmaxf(acc[t][v], 0.f);
    __syncthreads();

    // ---- layers 1,2: [16 x 64] @ [64 x 64] + b, relu ----
    const float* Ws[2] = {W1, W2};
    const float* bs[2] = {b1, b2};
    for (int L = 0; L < 2; ++L) {
        v8f a2[4];
        for (int t = 0; t < 4; ++t) {
            float bv = bs[L][t * 16 + colN];
            for (int v = 0; v < 8; ++v) a2[t][v] = bv;
        }
        for (int kk = 0; kk < MH; kk += 4) {
            int k0 = kk + koff;
            v2f a; a.x = hbuf[mrow][k0]; a.y = hbuf[mrow][k0 + 1];
            for (int t = 0; t < 4; ++t) {
                v2f bb;
                bb.x = Ws[L][k0 * MH + t * 16 + colN];
                bb.y = Ws[L][(k0 + 1) * MH + t * 16 + colN];
                a2[t] = __builtin_amdgcn_wmma_f32_16x16x4_f32(
                    false, a, false, bb, (short)0, a2[t], false, false);
            }
        }
        __syncthreads();
        for (int t = 0; t < 4; ++t)
            for (int v = 0; v < 8; ++v)
                hbuf[v + hi * 8][t * 16 + colN] = fmaxf(a2[t][v], 0.f);
        __syncthreads();
    }

    // ---- LayerNorm over 64 features, one lane per row ----
    if (lane < 16) {
        float mu = 0.f;
        for (int j = 0; j < MH; ++j) mu += hbuf[lane][j];
        mu *= (1.f / MH);
        float var = 0.f;
        for (int j = 0; j < MH; ++j) {
            float d = hbuf[lane][j] - mu;
            var += d * d;
        }
        var *= (1.f / MH);
        float inv = rsqrtf(var + LN_EPS);
        for (int j = 0; j < MH; ++j)
            hbuf[lane][j] = (hbuf[lane][j] - mu) * inv * lng[j] + lnb[j];
    }
    __syncthreads();

    // ---- layer 3: [16 x 64] @ [64 x 128] + b3, + u, store ----
    v8f o[8];
    for (int t = 0; t < 8; ++t) {
        float bv = b3[t * 16 + colN];
        for (int v = 0; v < 8; ++v) o[t][v] = bv;
    }
    for (int kk = 0; kk < MH; kk += 4) {
        int k0 = kk + koff;
        v2f a; a.x = hbuf[mrow][k0]; a.y = hbuf[mrow][k0 + 1];
        for (int t = 0; t < 8; ++t) {
            v2f bb;
            bb.x = W3[k0 * H + t * 16 + colN];
            bb.y = W3[(k0 + 1) * H + t * 16 + colN];
            o[t] = __builtin_amdgcn_wmma_f32_16x16x4_f32(
                false, a, false, bb, (short)0, o[t], false, false);
        }
    }
    for (int t = 0; t < 8; ++t)
        for (int v = 0; v < 8; ++v) {
            int r = v + hi * 8;
            size_t gi = (size_t)(rbase + r) * H + t * 16 + colN;
            out[gi] = o[t][v] + u[gi];
        }
}

// ---------------------------------------------------------------------------
// Launch
// ---------------------------------------------------------------------------
extern "C" void kernel_launch(void* const* d_in, const int* in_sizes, int n_in,
                              void* d_out, int out_size, void* d_ws, size_t ws_size,
                              hipStream_t stream) {
    const float* x     = (const float*)d_in[0];
    const float* u     = (const float*)d_in[3];
    const int*   batch = (const int*)d_in[4];
    const float* W0    = (const float*)d_in[5];
    const float* b0    = (const float*)d_in[6];
    const float* W1    = (const float*)d_in[7];
    const float* b1    = (const float*)d_in[8];
    const float* W2    = (const float*)d_in[9];
    const float* b2    = (const float*)d_in[10];
    const float* lng   = (const float*)d_in[11];
    const float* lnb   = (const float*)d_in[12];
    const float* W3    = (const float*)d_in[13];
    const float* b3    = (const float*)d_in[14];

    const int n_nodes = in_sizes[0] / H;   // 2,000,000
    const int nB      = in_sizes[3] / H;   // 1024

    float* ws  = (float*)d_ws;
    float* s   = ws;
    float* sq  = s + (size_t)nB * H;
    float* mi  = sq + (size_t)nB * H;
    float* ma  = mi + (size_t)nB * H;
    float* cnt = ma + (size_t)nB * H;
    float* cc  = cnt + nB;                 // concat [nB, 640]

    init_ws_kernel<<<(nB * H + 255) / 256, 256, 0, stream>>>(s, sq, mi, ma, cnt, nB);

    int nblk = (n_nodes + NPB - 1) / NPB;
    seg_reduce_kernel<<<nblk, 128, 0, stream>>>(x, batch, s, sq, mi, ma, cnt, n_nodes);

    build_concat_kernel<<<(nB * KC + 255) / 256, 256, 0, stream>>>(u, s, sq, mi, ma, cnt, cc, nB);

    mlp_kernel<<<nB / 16, 32, 0, stream>>>(cc, u, W0, b0, W1, b1, W2, b2,
                                           lng, lnb, W3, b3, (float*)d_out);
}